// INR_51814485459009
// MI455X (gfx1250) — compile-verified
//
#include <hip/hip_runtime.h>

typedef __attribute__((ext_vector_type(2))) float v2f;
typedef __attribute__((ext_vector_type(8))) float v8f;

#define HID     256
#define LDSROW  258            // padded row stride (floats): 8B-aligned b64 loads, conflict-free banks
#define WAVES   4
#define NPIX    (16 * 128 * 128)

static __device__ __forceinline__ v8f wmma_f32(v2f a, v2f b, v8f c) {
  return __builtin_amdgcn_wmma_f32_16x16x4_f32(false, a, false, b, (short)0, c, false, false);
}

// ---------------------------------------------------------------------------
// Weight pre-pack kernels: lay W out in WMMA B-fragment order so the hot loop
// does one coalesced global_load_b64 (256B/wave) per fragment.
// Fragment f holds, for lane L: { W[kb][col], W[kb+1][col] } with
// kb = k4*4 + (L>=16 ? 2 : 0), col = nt*16 + (L&15).
// Packed address (floats): f*64 + L*2, f = nt*64 + k4.
// ---------------------------------------------------------------------------
__global__ void pack_w256(const float* __restrict__ W, float* __restrict__ P) {
  const int idx = blockIdx.x * blockDim.x + threadIdx.x;   // 16*64*32 threads
  const int L   = idx & 31;
  const int f   = idx >> 5;
  const int k4  = f & 63;
  const int nt  = f >> 6;
  const int kb  = k4 * 4 + ((L >= 16) ? 2 : 0);
  const int col = nt * 16 + (L & 15);
  P[f * 64 + L * 2 + 0] = W[kb * HID + col];
  P[f * 64 + L * 2 + 1] = W[(kb + 1) * HID + col];
}

// Layer-1 fragments: B = [W1 row0; W1 row1; b1; 0]  (16 fragments, f = nt)
__global__ void pack_w1(const float* __restrict__ W1, const float* __restrict__ b1,
                        float* __restrict__ P) {
  const int idx = blockIdx.x * blockDim.x + threadIdx.x;   // 16*32 threads
  const int L   = idx & 31;
  const int nt  = idx >> 5;
  const int col = nt * 16 + (L & 15);
  const bool hi = L >= 16;
  P[nt * 64 + L * 2 + 0] = hi ? b1[col] : W1[col];
  P[nt * 64 + L * 2 + 1] = hi ? 0.0f    : W1[HID + col];
}

// Layer-4 fragments: 256x3 weights into one N-tile with cols >=3 zeroed (f = k4)
__global__ void pack_w4(const float* __restrict__ W4, float* __restrict__ P) {
  const int idx = blockIdx.x * blockDim.x + threadIdx.x;   // 64*32 threads
  const int L   = idx & 31;
  const int k4  = idx >> 5;
  const int m   = L & 15;
  const bool hi = L >= 16;
  const int kb  = k4 * 4 + (hi ? 2 : 0);
  const bool valid = m < 3;
  const int  mm    = valid ? m : 0;
  P[k4 * 64 + L * 2 + 0] = valid ? W4[kb * 3 + mm]       : 0.0f;
  P[k4 * 64 + L * 2 + 1] = valid ? W4[(kb + 1) * 3 + mm] : 0.0f;
}

// ---------------------------------------------------------------------------
// Packed dense 256->256 + ReLU, N register-blocked by 4:
// per K-step: 1 ds_load_b64 (A) + 4 global_load_b64 (B frags) + 4 v_wmma
// ---------------------------------------------------------------------------
static __device__ __forceinline__ void dense_relu_packed(const float* __restrict__ src,
                                                         float* __restrict__ dst,
                                                         const float* __restrict__ Wp,
                                                         const float* __restrict__ bg,
                                                         int m, bool hi, int lane) {
  for (int nt0 = 0; nt0 < 16; nt0 += 4) {
    v8f acc0 = {}, acc1 = {}, acc2 = {}, acc3 = {};
    const float* wbase = &Wp[(nt0 * 64) * 64 + lane * 2];
#pragma unroll 2
    for (int k4 = 0; k4 < HID / 4; ++k4) {
      v2f a  = *(const v2f*)&src[m * LDSROW + k4 * 4 + (hi ? 2 : 0)];
      const float* wf = wbase + k4 * 64;
      v2f bf0 = *(const v2f*)(wf);
      v2f bf1 = *(const v2f*)(wf + 1 * 64 * 64);
      v2f bf2 = *(const v2f*)(wf + 2 * 64 * 64);
      v2f bf3 = *(const v2f*)(wf + 3 * 64 * 64);
      acc0 = wmma_f32(a, bf0, acc0);
      acc1 = wmma_f32(a, bf1, acc1);
      acc2 = wmma_f32(a, bf2, acc2);
      acc3 = wmma_f32(a, bf3, acc3);
    }
    auto flush = [&](v8f acc, int nt) {
      const int col = nt * 16 + m;
      const float bias = bg[col];
#pragma unroll
      for (int r = 0; r < 8; ++r)
        dst[(r + (hi ? 8 : 0)) * LDSROW + col] = fmaxf(acc[r] + bias, 0.0f);
    };
    flush(acc0, nt0); flush(acc1, nt0 + 1); flush(acc2, nt0 + 2); flush(acc3, nt0 + 3);
  }
}

__global__ __launch_bounds__(32 * WAVES) void inr_mlp_packed(
    const float* __restrict__ x,
    const int*   __restrict__ sample_idx,
    const float* __restrict__ shift_vectors,
    const float* __restrict__ rotation_angle,
    const float* __restrict__ color_scales,
    const float* __restrict__ color_shifts,
    const float* __restrict__ w1p, const float* __restrict__ w2p,
    const float* __restrict__ w3p, const float* __restrict__ w4p,
    const float* __restrict__ b2,  const float* __restrict__ b3,
    const float* __restrict__ b4,
    float* __restrict__ out, float* __restrict__ dxdy)
{
  __shared__ float lds[WAVES * 2 * 16 * LDSROW];

  const int  lane = threadIdx.x & 31;
  const int  wave = threadIdx.x >> 5;
  const bool hi   = lane >= 16;
  const int  m    = lane & 15;

  float* buf0 = &lds[wave * (2 * 16 * LDSROW)];
  float* buf1 = buf0 + 16 * LDSROW;

  const int tile     = blockIdx.x * WAVES + wave;   // 16 pixels per wave
  const int pix_base = tile << 4;
  const int b        = pix_base >> 14;              // 128*128 pixels per image
  const int sidx     = sample_idx[b];

  const float ang = rotation_angle[sidx];
  const float cs  = cosf(ang);
  const float sn  = sinf(ang);
  const float dy  = shift_vectors[sidx * 2 + 0];
  const float dx  = shift_vectors[sidx * 2 + 1];

  // ---- Layer 1: A = [c0 c1 1 0] (bias folded into K), B fragments pre-packed
  v2f afrag;
  {
    const float x0 = x[(pix_base + m) * 2 + 0];
    const float x1 = x[(pix_base + m) * 2 + 1];
    const float c0 = cs * x0 - sn * x1 + dx;
    const float c1 = sn * x0 + cs * x1 + dy;
    afrag.x = hi ? 1.0f : c0;
    afrag.y = hi ? 0.0f : c1;
  }

#pragma unroll 4
  for (int nt = 0; nt < 16; ++nt) {
    v2f bfrag = *(const v2f*)&w1p[nt * 64 + lane * 2];
    v8f acc = {};
    acc = wmma_f32(afrag, bfrag, acc);
    const int col = nt * 16 + m;
#pragma unroll
    for (int r = 0; r < 8; ++r)
      buf0[(r + (hi ? 8 : 0)) * LDSROW + col] = fmaxf(acc[r], 0.0f);
  }
  __syncthreads();

  dense_relu_packed(buf0, buf1, w2p, b2, m, hi, lane);   // Layer 2
  __syncthreads();
  dense_relu_packed(buf1, buf0, w3p, b3, m, hi, lane);   // Layer 3
  __syncthreads();

  // ---- Layer 4: 256 -> 3 via one masked N-tile + color epilogue
  {
    v8f acc = {};
#pragma unroll 2
    for (int k4 = 0; k4 < HID / 4; ++k4) {
      v2f a  = *(const v2f*)&buf0[m * LDSROW + k4 * 4 + (hi ? 2 : 0)];
      v2f bf = *(const v2f*)&w4p[k4 * 64 + lane * 2];
      acc = wmma_f32(a, bf, acc);
    }
    const bool valid = m < 3;
    float bias = 0.0f, scl = 1.0f, sft = 0.0f;
    if (valid) {
      bias = b4[m];
      if (sidx != 0) {
        scl = color_scales[sidx * 3 + m];
        sft = color_shifts[sidx * 3 + m];
      }
    }
#pragma unroll
    for (int r = 0; r < 8; ++r) {
      const int row = r + (hi ? 8 : 0);
      if (valid)
        out[(pix_base + row) * 3 + m] = (acc[r] + bias) * scl + sft;
    }
  }

  if (blockIdx.x == 0 && threadIdx.x < 16) {
    const int bb = threadIdx.x;
    const int si = sample_idx[bb];
    dxdy[bb]      = shift_vectors[si * 2 + 1];  // dx
    dxdy[16 + bb] = shift_vectors[si * 2 + 0];  // dy
  }
}

// ---------------------------------------------------------------------------
// Fallback (no scratch): unpacked weights, 2x global_load_b32 per WMMA
// ---------------------------------------------------------------------------
static __device__ __forceinline__ void dense_relu(const float* __restrict__ src,
                                                  float* __restrict__ dst,
                                                  const float* __restrict__ Wg,
                                                  const float* __restrict__ bg,
                                                  int m, bool hi) {
  for (int nt = 0; nt < 16; ++nt) {
    const int col = nt * 16 + m;
    v8f acc = {};
#pragma unroll 4
    for (int k4 = 0; k4 < HID / 4; ++k4) {
      const int kb = k4 * 4 + (hi ? 2 : 0);
      v2f a = *(const v2f*)&src[m * LDSROW + kb];
      v2f bf;
      bf.x = Wg[kb * HID + col];
      bf.y = Wg[(kb + 1) * HID + col];
      acc = wmma_f32(a, bf, acc);
    }
    const float bias = bg[col];
#pragma unroll
    for (int r = 0; r < 8; ++r)
      dst[(r + (hi ? 8 : 0)) * LDSROW + col] = fmaxf(acc[r] + bias, 0.0f);
  }
}

__global__ __launch_bounds__(32 * WAVES) void inr_mlp_fallback(
    const float* __restrict__ x,
    const int*   __restrict__ sample_idx,
    const float* __restrict__ shift_vectors,
    const float* __restrict__ rotation_angle,
    const float* __restrict__ color_scales,
    const float* __restrict__ color_shifts,
    const float* __restrict__ W1, const float* __restrict__ b1,
    const float* __restrict__ W2, const float* __restrict__ b2,
    const float* __restrict__ W3, const float* __restrict__ b3,
    const float* __restrict__ W4, const float* __restrict__ b4,
    float* __restrict__ out, float* __restrict__ dxdy)
{
  __shared__ float lds[WAVES * 2 * 16 * LDSROW];

  const int  lane = threadIdx.x & 31;
  const int  wave = threadIdx.x >> 5;
  const bool hi   = lane >= 16;
  const int  m    = lane & 15;

  float* buf0 = &lds[wave * (2 * 16 * LDSROW)];
  float* buf1 = buf0 + 16 * LDSROW;

  const int tile     = blockIdx.x * WAVES + wave;
  const int pix_base = tile << 4;
  const int b        = pix_base >> 14;
  const int sidx     = sample_idx[b];

  const float ang = rotation_angle[sidx];
  const float cs  = cosf(ang);
  const float sn  = sinf(ang);
  const float dy  = shift_vectors[sidx * 2 + 0];
  const float dx  = shift_vectors[sidx * 2 + 1];

  v2f afrag;
  {
    const float x0 = x[(pix_base + m) * 2 + 0];
    const float x1 = x[(pix_base + m) * 2 + 1];
    const float c0 = cs * x0 - sn * x1 + dx;
    const float c1 = sn * x0 + cs * x1 + dy;
    afrag.x = hi ? 1.0f : c0;
    afrag.y = hi ? 0.0f : c1;
  }

  for (int nt = 0; nt < 16; ++nt) {
    const int col = nt * 16 + m;
    v2f bfrag;
    if (!hi) { bfrag.x = W1[col];  bfrag.y = W1[HID + col]; }
    else     { bfrag.x = b1[col];  bfrag.y = 0.0f; }
    v8f acc = {};
    acc = wmma_f32(afrag, bfrag, acc);
#pragma unroll
    for (int r = 0; r < 8; ++r)
      buf0[(r + (hi ? 8 : 0)) * LDSROW + col] = fmaxf(acc[r], 0.0f);
  }
  __syncthreads();

  dense_relu(buf0, buf1, W2, b2, m, hi);
  __syncthreads();
  dense_relu(buf1, buf0, W3, b3, m, hi);
  __syncthreads();

  {
    const bool valid = m < 3;
    const int  mm    = valid ? m : 2;
    v8f acc = {};
#pragma unroll 4
    for (int k4 = 0; k4 < HID / 4; ++k4) {
      const int kb = k4 * 4 + (hi ? 2 : 0);
      v2f a = *(const v2f*)&buf0[m * LDSROW + kb];
      const float w0 = W4[kb * 3 + mm];
      const float w1 = W4[(kb + 1) * 3 + mm];
      v2f bf;
      bf.x = valid ? w0 : 0.0f;
      bf.y = valid ? w1 : 0.0f;
      acc = wmma_f32(a, bf, acc);
    }
    float bias = 0.0f, scl = 1.0f, sft = 0.0f;
    if (valid) {
      bias = b4[m];
      if (sidx != 0) {
        scl = color_scales[sidx * 3 + m];
        sft = color_shifts[sidx * 3 + m];
      }
    }
#pragma unroll
    for (int r = 0; r < 8; ++r) {
      const int row = r + (hi ? 8 : 0);
      if (valid)
        out[(pix_base + row) * 3 + m] = (acc[r] + bias) * scl + sft;
    }
  }

  if (blockIdx.x == 0 && threadIdx.x < 16) {
    const int bb = threadIdx.x;
    const int si = sample_idx[bb];
    dxdy[bb]      = shift_vectors[si * 2 + 1];
    dxdy[16 + bb] = shift_vectors[si * 2 + 0];
  }
}

extern "C" void kernel_launch(void* const* d_in, const int* in_sizes, int n_in,
                              void* d_out, int out_size, void* d_ws, size_t ws_size,
                              hipStream_t stream) {
  const float* x              = (const float*)d_in[0];
  const int*   sample_idx     = (const int*)  d_in[1];
  const float* shift_vectors  = (const float*)d_in[2];
  const float* rotation_angle = (const float*)d_in[3];
  const float* color_scales   = (const float*)d_in[4];
  const float* color_shifts   = (const float*)d_in[5];
  const float* W1 = (const float*)d_in[6];
  const float* b1 = (const float*)d_in[7];
  const float* W2 = (const float*)d_in[8];
  const float* b2 = (const float*)d_in[9];
  const float* W3 = (const float*)d_in[10];
  const float* b3 = (const float*)d_in[11];
  const float* W4 = (const float*)d_in[12];
  const float* b4 = (const float*)d_in[13];

  float* out  = (float*)d_out;
  float* dxdy = out + (size_t)NPIX * 3;

  const int tiles  = NPIX / 16;
  const int blocks = tiles / WAVES;

  // Packed-fragment scratch layout (floats):
  //   w2p[65536] | w3p[65536] | w1p[1024] | w4p[4096]   -> 544768 bytes
  const size_t need_floats = 65536 + 65536 + 1024 + 4096;
  if (ws_size >= need_floats * sizeof(float)) {
    float* wsf = (float*)d_ws;
    float* w2p = wsf;
    float* w3p = wsf + 65536;
    float* w1p = wsf + 131072;
    float* w4p = wsf + 132096;

    pack_w256<<<(16 * 64 * 32) / 256, 256, 0, stream>>>(W2, w2p);
    pack_w256<<<(16 * 64 * 32) / 256, 256, 0, stream>>>(W3, w3p);
    pack_w1  <<<2, 256, 0, stream>>>(W1, b1, w1p);
    pack_w4  <<<8, 256, 0, stream>>>(W4, w4p);

    inr_mlp_packed<<<blocks, 32 * WAVES, 0, stream>>>(
        x, sample_idx, shift_vectors, rotation_angle, color_scales, color_shifts,
        w1p, w2p, w3p, w4p, b2, b3, b4, out, dxdy);
  } else {
    inr_mlp_fallback<<<blocks, 32 * WAVES, 0, stream>>>(
        x, sample_idx, shift_vectors, rotation_angle, color_scales, color_shifts,
        W1, b1, W2, b2, W3, b3, W4, b4, out, dxdy);
  }
}